// ContextSegmentScoringModule_3040836846436
// MI455X (gfx1250) — compile-verified
//
#include <hip/hip_runtime.h>

typedef __attribute__((ext_vector_type(2))) float v2f;
typedef __attribute__((ext_vector_type(4))) float v4f;
typedef __attribute__((ext_vector_type(8))) float v8f;

// Problem constants from the reference: B=32, N=32*32*8=8192, E=256.
constexpr int E_DIM          = 256;
constexpr int ROWS_PER_BATCH = 8192;

// One wave computes 16 output dots (one 16-row tile of ctx against the
// batch's segment vector) via V_WMMA_F32_16X16X4_F32, accumulating over
// E in 64 steps of K=4.
__global__ __launch_bounds__(256) void ctxseg_wmma_kernel(
    const float* __restrict__ ctx,   // [B*N, E] row-major
    const float* __restrict__ seg,   // [B, E]
    float* __restrict__ out)         // [B*N]
{
    __shared__ float sseg[E_DIM];

    const int lane      = threadIdx.x & 31;
    const int waveInBlk = threadIdx.x >> 5;
    const size_t r0 = ((size_t)blockIdx.x * (blockDim.x >> 5) + waveInBlk) * 16;

    // All 128 rows covered by this block live in one batch (8192 % 128 == 0).
    const int batch = (int)(r0 / ROWS_PER_BATCH);

    // Stage the 1 KB segment embedding into LDS once per block.
    sseg[threadIdx.x] = seg[(size_t)batch * E_DIM + threadIdx.x];
    __syncthreads();

    // A-fragment layout (16x4 f32, 2 VGPRs):
    //   lane L: row M = L&15; lanes 0-15 hold K={kk,kk+1}, lanes 16-31 K={kk+2,kk+3}
    const int m    = lane & 15;
    const int half = lane >> 4;

    const float* aptr = ctx + (r0 + (size_t)m) * E_DIM + 2 * half;

    v8f acc = {};
#pragma unroll 8
    for (int kk = 0; kk < E_DIM; kk += 4) {
        // A: 8B per lane; lane m & lane m+16 form a contiguous 16B span of row m.
        v2f a = *(const v2f*)(aptr + kk);

        // B-fragment layout (4x16 f32, 2 VGPRs): VGPR0 = rows K=0/K=1,
        // VGPR1 = rows K=2/K=3 (split by lane half). Broadcast the segment
        // chunk to every N-column -> all 16 D columns equal the desired dots.
        v2f b;
        b.x = sseg[kk + half];
        b.y = sseg[kk + 2 + half];

        // (neg_a, A, neg_b, B, c_mod, C, reuse_a, reuse_b)
        acc = __builtin_amdgcn_wmma_f32_16x16x4_f32(
            false, a, false, b, (short)0, acc, false, false);
    }

    // D layout: lane n, VGPR v -> D(M = v + 8*(n>=16), n). Columns identical;
    // take column 0: lane 0 stores rows 0-7, lane 16 stores rows 8-15.
    if (m == 0) {
        float* o = out + r0 + 8 * half;
        v4f lo = { acc[0], acc[1], acc[2], acc[3] };
        v4f hi = { acc[4], acc[5], acc[6], acc[7] };
        *(v4f*)(o)     = lo;
        *(v4f*)(o + 4) = hi;
    }
}

extern "C" void kernel_launch(void* const* d_in, const int* in_sizes, int n_in,
                              void* d_out, int out_size, void* d_ws, size_t ws_size,
                              hipStream_t stream) {
    const float* ctx = (const float*)d_in[0];   // c_embeds [32,32,32,8,256] f32
    const float* seg = (const float*)d_in[1];   // s_embeds [32,256] f32
    float* out = (float*)d_out;                 // [32, 8192] f32

    const int total_rows = out_size;            // 262144
    const int waves      = total_rows / 16;     // 16384 tiles
    const int threads    = 256;                 // 8 waves/block
    const int blocks     = waves / (threads / 32); // 2048
    ctxseg_wmma_kernel<<<blocks, threads, 0, stream>>>(ctx, seg, out);
}